// GNN_56822417326534
// MI455X (gfx1250) — compile-verified
//
#include <hip/hip_runtime.h>
#include <math.h>

#define N_NODES 200000
#define N_EDGES 2400000
#define N_GRAPH 128

typedef __attribute__((ext_vector_type(2)))  float    v2f;
typedef __attribute__((ext_vector_type(8)))  float    v8f;

// Hardware f32 atomic add (global_atomic_add_f32), not a CAS loop.
__device__ __forceinline__ void atomAddF(float* p, float v) {
    unsafeAtomicAdd(p, v);
}

// ---------------- zero scratch ----------------
__global__ void k_zero(float* __restrict__ p, int n) {
    int i = blockIdx.x * blockDim.x + threadIdx.x;
    int stride = gridDim.x * blockDim.x;
    for (; i < n; i += stride) p[i] = 0.0f;
}

// ---------------- degree: deg[dst] += 1 ----------------
__global__ void k_degree(const int* __restrict__ ei, float* __restrict__ deg) {
    int e = blockIdx.x * blockDim.x + threadIdx.x;
    if (e >= N_EDGES) return;
    atomAddF(&deg[ei[N_EDGES + e]], 1.0f);
}

// ---------------- node pass 1: norm coefs + h1 = x @ W1 ----------------
// deg accumulated into `dinv` slot; overwritten in place.
__global__ void k_node1(const float* __restrict__ x, const float* __restrict__ W1,
                        float* __restrict__ dinvs, float* __restrict__ dinv,
                        float* __restrict__ h1) {
    int n = blockIdx.x * blockDim.x + threadIdx.x;
    if (n >= N_NODES) return;
    float d = dinv[n] + 1.0f;            // self-loop
    dinvs[n] = rsqrtf(d);
    dinv[n]  = 1.0f / d;
    float x0 = x[2 * n], x1 = x[2 * n + 1];
#pragma unroll
    for (int j = 0; j < 16; ++j)
        h1[n * 16 + j] = x0 * W1[j] + x1 * W1[16 + j];
}

// ---------------- edge scatter, 16 feats (16 lanes per edge) ----------------
__global__ void k_edge16(const int* __restrict__ ei, const float* __restrict__ dinvs,
                         const float* __restrict__ h1, float* __restrict__ agg1) {
    int t = blockIdx.x * blockDim.x + threadIdx.x;   // E*16 = 38.4M threads
    int e = t >> 4, f = t & 15;
    if (e >= N_EDGES) return;
    int s = ei[e], d = ei[N_EDGES + e];
    float coef = dinvs[s] * dinvs[d];
    atomAddF(&agg1[d * 16 + f], h1[s * 16 + f] * coef);
}

// ---------------- edge scatter, 32 feats (32 lanes per edge) ----------------
__global__ void k_edge32(const int* __restrict__ ei, const float* __restrict__ dinvs,
                         const float* __restrict__ h2, float* __restrict__ agg2) {
    int t = blockIdx.x * blockDim.x + threadIdx.x;   // E*32 = 76.8M threads
    int e = t >> 5, f = t & 31;
    if (e >= N_EDGES) return;
    int s = ei[e], d = ei[N_EDGES + e];
    float coef = dinvs[s] * dinvs[d];
    atomAddF(&agg2[d * 32 + f], h2[s * 32 + f] * coef);
}

// ---------------- layer-2: relu(agg1 + h1*dinv + b1) @ W2 via WMMA ----------------
// One wave per 16-node tile. A = relu tile [16x16], B = W2 [16x32] (two 16-wide halves).
// Exact-f32 WMMA: D[16x16] += A[16x4] x B[4x16], 4 K-chunks x 2 N-halves.
// Each lane loads ONLY the 8 A-elements it owns (lane-varying column = VMEM
// gather) so no dynamic register indexing / cndmask chains are generated.
__global__ void __launch_bounds__(256)
k_layer2(const float* __restrict__ agg1, const float* __restrict__ h1,
         const float* __restrict__ dinv, const float* __restrict__ b1,
         const float* __restrict__ W2, float* __restrict__ h2) {
    int wave = threadIdx.x >> 5;
    int lane = threadIdx.x & 31;
    int tile = blockIdx.x * 8 + wave;
    if (tile >= (N_NODES / 16)) return;   // wave-granular exit: EXEC all-1 inside
    int base = tile * 16;

    bool hi  = lane >= 16;
    int  col = lane & 15;
    int  c0i = hi ? 2 : 0;                // A/B K-offsets this lane feeds
    int  c1i = hi ? 3 : 1;

    // A layout (ISA 7.12.2, 32-bit A 16x4): lanes 0-15 hold K={4k,4k+1},
    // lanes 16-31 hold K={4k+2,4k+3}; M = lane&15.
    int node = base + col;
    float dv = dinv[node];
    const float* ar = agg1 + node * 16;
    const float* hr = h1   + node * 16;

    v8f c0 = {0.f,0.f,0.f,0.f,0.f,0.f,0.f,0.f};
    v8f c1 = c0;
#pragma unroll
    for (int k = 0; k < 4; ++k) {
        int j0 = 4 * k + c0i;
        int j1 = 4 * k + c1i;
        float a0 = ar[j0] + hr[j0] * dv + b1[j0];  a0 = a0 > 0.0f ? a0 : 0.0f;
        float a1 = ar[j1] + hr[j1] * dv + b1[j1];  a1 = a1 > 0.0f ? a1 : 0.0f;
        v2f a;   a.x   = a0;                   a.y   = a1;
        v2f b0;  b0.x  = W2[j0 * 32 + col];    b0.y  = W2[j1 * 32 + col];
        v2f b1v; b1v.x = W2[j0 * 32 + 16 + col]; b1v.y = W2[j1 * 32 + 16 + col];
        c0 = __builtin_amdgcn_wmma_f32_16x16x4_f32(false, a, false, b0,  (short)0, c0, false, false);
        c1 = __builtin_amdgcn_wmma_f32_16x16x4_f32(false, a, false, b1v, (short)0, c1, false, false);
    }

    // D layout: VGPR r -> M=r (lanes 0-15) / M=8+r (lanes 16-31), N=lane&15
#pragma unroll
    for (int r = 0; r < 8; ++r) {
        int m = hi ? (8 + r) : r;
        h2[(base + m) * 32 + col]      = c0[r];
        h2[(base + m) * 32 + 16 + col] = c1[r];
    }
}

// ---------------- pooling: streaming segment-sum over sorted batch ----------------
// One wave per 128-node chunk; lane = feature. Register-accumulate, flush on
// graph boundary -> ~2 atomics per wave instead of 32 per node.
__global__ void k_pool(const float* __restrict__ agg2, const float* __restrict__ h2,
                       const float* __restrict__ dinv, const float* __restrict__ b2,
                       const int* __restrict__ batch, float* __restrict__ pools,
                       float* __restrict__ poolc) {
    const int CHUNK = 128;
    int lane = threadIdx.x;
    int n0 = blockIdx.x * CHUNK;
    if (n0 >= N_NODES) return;
    int n1 = n0 + CHUNK; if (n1 > N_NODES) n1 = N_NODES;
    float bb = b2[lane];
    float acc = 0.0f, cnt = 0.0f;
    int gcur = batch[n0];
    for (int n = n0; n < n1; ++n) {
        int g = batch[n];                 // uniform across wave
        if (g != gcur) {
            atomAddF(&pools[gcur * 32 + lane], acc);
            if (lane == 0) atomAddF(&poolc[gcur], cnt);
            acc = 0.0f; cnt = 0.0f; gcur = g;
        }
        acc += agg2[n * 32 + lane] + h2[n * 32 + lane] * dinv[n] + bb;
        cnt += 1.0f;
    }
    atomAddF(&pools[gcur * 32 + lane], acc);
    if (lane == 0) atomAddF(&poolc[gcur], cnt);
}

// ---------------- head: (sum/cnt) @ Wl + bl, sigmoid ----------------
__global__ void k_out(const float* __restrict__ pools, const float* __restrict__ poolc,
                      const float* __restrict__ Wl, const float* __restrict__ bl,
                      float* __restrict__ out) {
    int g = threadIdx.x;
    if (g >= N_GRAPH) return;
    float cnt = poolc[g]; if (cnt < 1.0f) cnt = 1.0f;
    float s = 0.0f;
#pragma unroll
    for (int j = 0; j < 32; ++j) s += pools[g * 32 + j] * Wl[j];
    s = s / cnt + bl[0];
    out[g] = 1.0f / (1.0f + expf(-s));
}

extern "C" void kernel_launch(void* const* d_in, const int* in_sizes, int n_in,
                              void* d_out, int out_size, void* d_ws, size_t ws_size,
                              hipStream_t stream) {
    const float* x   = (const float*)d_in[0];
    const int*   ei  = (const int*)  d_in[1];
    const int*   bat = (const int*)  d_in[2];
    const float* W1  = (const float*)d_in[3];
    const float* b1  = (const float*)d_in[4];
    const float* W2  = (const float*)d_in[5];
    const float* b2  = (const float*)d_in[6];
    const float* Wl  = (const float*)d_in[7];
    const float* bl  = (const float*)d_in[8];

    float* ws    = (float*)d_ws;              // 98*N + 33*G floats (~78.4 MB)
    float* dinvs = ws;                        // [N]
    float* dinv  = ws + 1  * N_NODES;         // [N]  (deg accumulator first)
    float* h1    = ws + 2  * N_NODES;         // [N,16]
    float* agg1  = ws + 18 * N_NODES;         // [N,16]
    float* h2    = ws + 34 * N_NODES;         // [N,32]
    float* agg2  = ws + 66 * N_NODES;         // [N,32]
    float* pools = ws + 98 * N_NODES;         // [G,32]
    float* poolc = pools + 32 * N_GRAPH;      // [G]

    k_zero<<<2048, 256, 0, stream>>>(dinv, N_NODES);
    k_zero<<<2048, 256, 0, stream>>>(agg1, 16 * N_NODES);
    k_zero<<<4096, 256, 0, stream>>>(agg2, 32 * N_NODES);
    k_zero<<<32,   256, 0, stream>>>(pools, 33 * N_GRAPH);

    k_degree<<<(N_EDGES + 255) / 256, 256, 0, stream>>>(ei, dinv);
    k_node1 <<<(N_NODES + 255) / 256, 256, 0, stream>>>(x, W1, dinvs, dinv, h1);
    k_edge16<<<(N_EDGES * 16) / 256,  256, 0, stream>>>(ei, dinvs, h1, agg1);
    k_layer2<<<((N_NODES / 16) + 7) / 8, 256, 0, stream>>>(agg1, h1, dinv, b1, W2, h2);
    k_edge32<<<(N_EDGES * 32) / 256,  256, 0, stream>>>(ei, dinvs, h2, agg2);
    k_pool  <<<(N_NODES + 127) / 128, 32, 0, stream>>>(agg2, h2, dinv, b2, bat, pools, poolc);
    k_out   <<<1, 128, 0, stream>>>(pools, poolc, Wl, bl, (float*)d_out);
}